// PolyphaseDiff_17506286698984
// MI455X (gfx1250) — compile-verified
//
#include <hip/hip_runtime.h>

// Polyphase resampler (up=3, down=2), N=2097152, 129-tap FIR, fp32.
// Formulated as banded-Toeplitz GEMM on V_WMMA_F32_16X16X4_F32 (fp32 WMMA,
// full precision). 4096 blocks x 96 threads (3 waves = 3 phases).

#define N_SIG   2097152
#define L_FILT  129
#define N_OUT   3145728
#define TILE_OUT 768          // outputs per block (divides N_OUT: 4096 tiles)
#define NBLOCKS  4096
#define WIN      560          // LDS x-window (needs 557)
#define NTHREADS 96
#define KSTEPS   19           // 19 * K4 = 76 >= band span 73

typedef float v2f __attribute__((ext_vector_type(2)));
typedef float v8f __attribute__((ext_vector_type(8)));

__global__ __launch_bounds__(NTHREADS)
void resample_poly_wmma(const float* __restrict__ x,
                        const float* __restrict__ h,
                        float* __restrict__ out)
{
    __shared__ float xw[WIN];                    // staged input window
    __shared__ float hs[L_FILT];                 // staged filter taps
    __shared__ __align__(16) float ob[TILE_OUT]; // output staging for coalesced store

    const int tile = blockIdx.x;
    const int tid  = threadIdx.x;
    const int I0   = tile * 256;      // first phase-index i of this tile
    const int W    = 2 * I0 - 21;     // global x index of xw[0]

    // Stage x window into LDS, zero-padded at signal edges (matches 'full' conv).
    for (int j = tid; j < WIN; j += NTHREADS) {
        int gi = W + j;
        xw[j] = (gi >= 0 && gi < N_SIG) ? x[gi] : 0.0f;
    }
    // Stage filter taps (tiny, stays hot).
    for (int j = tid; j < L_FILT; j += NTHREADS) hs[j] = h[j];
    __syncthreads();

    const int wave = tid >> 5;          // wave == phase p in {0,1,2}
    const int lane = tid & 31;
    const int p    = wave;
    const int d0   = (p == 0) ? -21 : -20;   // first tap offset d for this phase
    const int basep= (p == 0) ? 0 : 1;       // xw offset so W+basep == 2*I0+d0

    const int r  = lane & 15;   // A-matrix row (lanes 0-15 and 16-31 both hold M=0..15)
    const int hi = lane >> 4;   // 0 -> K pair {0,1}; 1 -> K pair {2,3}
    const int n  = lane & 15;   // B/D column

    // Build the banded filter matrix A[r][k] = g_p[k-2r] (g_p[t] = 3*h[64+2p-3*(d0+t)])
    // as 19 per-lane v2f fragments. Constant for the whole tile -> registers.
    v2f a[KSTEPS];
#pragma unroll
    for (int s = 0; s < KSTEPS; ++s) {
        const int k0 = 4 * s + 2 * hi;
        const int t0 = k0 - 2 * r;
        float ax = 0.0f, ay = 0.0f;
        if (t0 >= 0 && t0 < 43)         ax = 3.0f * hs[64 + 2 * p - 3 * (d0 + t0)];
        if (t0 + 1 >= 0 && t0 + 1 < 43) ay = 3.0f * hs[64 + 2 * p - 3 * (d0 + t0 + 1)];
        v2f av; av.x = ax; av.y = ay;
        a[s] = av;
    }

    // D = A x B accumulation over 19 fp32 WMMA steps (K=4 each).
    v8f acc = {0.f, 0.f, 0.f, 0.f, 0.f, 0.f, 0.f, 0.f};
    const int bbase = basep + 32 * n + 2 * hi;   // max 483 + 73 = 556 < WIN
#pragma unroll
    for (int s = 0; s < KSTEPS; ++s) {
        v2f b;
        b.x = xw[bbase + 4 * s];
        b.y = xw[bbase + 4 * s + 1];
        acc = __builtin_amdgcn_wmma_f32_16x16x4_f32(
            /*neg_a=*/false, a[s], /*neg_b=*/false, b,
            /*c_mod=*/(short)0, acc, /*reuse_a=*/false, /*reuse_b=*/false);
    }

    // D(M=rbase+v, N=n) = out[3*(I0 + 16n + M) + p]; scatter into LDS chunk.
    const int rbase = hi * 8;
#pragma unroll
    for (int v = 0; v < 8; ++v) {
        ob[3 * (16 * n + rbase + v) + p] = acc[v];
    }
    __syncthreads();

    // Coalesced vectorized store of the contiguous 768-output chunk.
    float4*       out4 = (float4*)(out + tile * TILE_OUT);
    const float4* ob4  = (const float4*)ob;
    out4[tid]      = ob4[tid];
    out4[tid + 96] = ob4[tid + 96];
}

extern "C" void kernel_launch(void* const* d_in, const int* in_sizes, int n_in,
                              void* d_out, int out_size, void* d_ws, size_t ws_size,
                              hipStream_t stream) {
    (void)in_sizes; (void)n_in; (void)d_ws; (void)ws_size; (void)out_size;
    const float* x = (const float*)d_in[0];
    const float* h = (const float*)d_in[1];
    // d_in[2]=up(3), d_in[3]=down(2): fixed by the reference configuration.
    float* out = (float*)d_out;
    resample_poly_wmma<<<NBLOCKS, NTHREADS, 0, stream>>>(x, h, out);
}